// Soft2FramesRecModule_84241488544396
// MI455X (gfx1250) — compile-verified
//
#include <hip/hip_runtime.h>

#define AS1 __attribute__((address_space(1)))
#define AS3 __attribute__((address_space(3)))

typedef float v4f __attribute__((ext_vector_type(4)));
typedef int   v4i __attribute__((ext_vector_type(4)));

// Problem constants (match the reference).
constexpr int S  = 16;    // leaf segments
constexpr int B  = 64;    // batch
constexpr int T  = 512;   // timesteps
constexpr int NZ = 512;   // encoding width
constexpr int TT = 64;    // timesteps per block
constexpr int C4 = NZ / 4;      // 128 float4 per row
constexpr int THREADS = 256;    // 8 wave32 waves

// ---- CDNA5 async global->LDS copy (ASYNCcnt path) ----------------------
__device__ __forceinline__ void async_copy_b128(const float* gsrc, v4f* ldst) {
#if __has_builtin(__builtin_amdgcn_global_load_async_to_lds_b128)
  __builtin_amdgcn_global_load_async_to_lds_b128((AS1 v4i*)gsrc, (AS3 v4i*)ldst,
                                                 /*offset=*/0, /*cpol=*/0);
#else
  unsigned laddr = (unsigned)(unsigned long long)(AS3 void*)ldst;
  asm volatile("global_load_async_to_lds_b128 %0, %1, off"
               :: "v"(laddr), "v"(gsrc) : "memory");
#endif
}

__device__ __forceinline__ void wait_async0() {
#if __has_builtin(__builtin_amdgcn_s_wait_asynccnt)
  __builtin_amdgcn_s_wait_asynccnt(0);
#else
  asm volatile("s_wait_asynccnt 0" ::: "memory");
#endif
}

// Non-temporal 128-bit store: 201 MB of output > 192 MB L2, never re-read.
__device__ __forceinline__ void nt_store(float* p, v4f v) {
  __builtin_nontemporal_store(v, (v4f*)p);
}

__global__ __launch_bounds__(THREADS)
void seg_interp_kernel(const float* __restrict__ seg_start,
                       const float* __restrict__ seg_end,
                       const float* __restrict__ e0,
                       const float* __restrict__ eg,
                       float* __restrict__ out) {
  // Exactly 64 KB static LDS: all 16 segment rows of e0/eg for this batch elem.
  __shared__ v4f lds_e0[S * C4];   // 32 KB
  __shared__ v4f lds_eg[S * C4];   // 32 KB

  const int b   = blockIdx.x / (T / TT);
  const int t0  = (blockIdx.x % (T / TT)) * TT;
  const int tid = threadIdx.x;

  // ---- Stage e0/eg rows for this b into LDS (async b128, 16 per thread) ----
  for (int i = tid; i < S * C4; i += THREADS) {
    const int s = i / C4, c = i % C4;
    const size_t goff = ((size_t)(s * B + b)) * NZ + (size_t)c * 4;
    async_copy_b128(e0 + goff, &lds_e0[i]);
    async_copy_b128(eg + goff, &lds_eg[i]);
  }

  // Block-uniform segment bounds -> scalar loads / SGPRs.
  float sv[S], ev[S];
#pragma unroll
  for (int s = 0; s < S; ++s) {
    sv[s] = seg_start[s * B + b];
    ev[s] = seg_end[s * B + b];
  }

  // ---- reset_indicator: each (b,t) owned by exactly one thread ----
  if (tid < TT) {
    const int t = t0 + tid;
    float ind = 0.0f;
#pragma unroll
    for (int s = 0; s < S; ++s) {
      int cs = (int)ceilf(sv[s]);
      cs = cs < 0 ? 0 : (cs > T - 1 ? T - 1 : cs);
      if (cs == t) ind = 1.0f;
    }
    out[(size_t)B * T * NZ + (size_t)B * T * 2 * NZ + (size_t)b * T + t] = ind;
  }

  wait_async0();
  __syncthreads();

  // ---- Main streaming loop: 64 t-rows x 128 float4 columns ----
  for (int idx = tid; idx < TT * C4; idx += THREADS) {
    const int tl = idx / C4, c = idx % C4;
    const int t  = t0 + tl;

    // Last segment covering t wins (matches sequential scatter semantics).
    int sel = -1;
#pragma unroll
    for (int s = 0; s < S; ++s) {
      const int cs = (int)ceilf(sv[s]);
      const int fe = (int)floorf(ev[s]);
      if (t >= cs && t <= fe) sel = s;
    }
    const float cov = (sel >= 0) ? 1.0f : 0.0f;
    const int   ss  = (sel >= 0) ? sel : 0;

    const float st    = sv[ss] * cov;
    const float en    = ev[ss] * cov;
    const float inter = ((float)t - st) / (en - st + 1e-7f);

    v4f z = lds_e0[ss * C4 + c];   // ds_load_b128
    v4f a = lds_eg[ss * C4 + c];
    z = z * cov;
    a = a * cov;
    const v4f cell = (a - z) * inter + z;   // == 0 when uncovered

    const size_t cb = ((size_t)b * T + t) * NZ + (size_t)c * 4;
    nt_store(out + cb, cell);

    const size_t rb = (size_t)B * T * NZ + ((size_t)b * T + t) * (size_t)(2 * NZ)
                    + (size_t)c * 4;
    nt_store(out + rb, a);        // reset_input[..., :NZ]  = e_gs
    nt_store(out + rb + NZ, z);   // reset_input[..., NZ:]  = e_0s
  }
}

extern "C" void kernel_launch(void* const* d_in, const int* in_sizes, int n_in,
                              void* d_out, int out_size, void* d_ws, size_t ws_size,
                              hipStream_t stream) {
  const float* seg_start = (const float*)d_in[0];
  const float* seg_end   = (const float*)d_in[1];
  const float* e0        = (const float*)d_in[2];
  const float* eg        = (const float*)d_in[3];
  float* out = (float*)d_out;

  dim3 grid(B * (T / TT));   // 512 blocks, one (b, t-tile) each
  seg_interp_kernel<<<grid, THREADS, 0, stream>>>(seg_start, seg_end, e0, eg, out);
}